// JointNet_23648089931873
// MI455X (gfx1250) — compile-verified
//
#include <hip/hip_runtime.h>

// RNN-T joint network, decomposed:
//   TW[m,v] = trans_2d[m,:] . W[v,:] + bias[v]   (m = b*T+t, 3200 rows)
//   PW[m,v] = preds_2d[m,:] . W[v,:]             (m = b*U1+u, 808 rows)
//   out[b,t,u,v] = leaky( TW[b*T+t, v] + PW[b*U1+u, v] )
// Compute drops from 8.27 GFLOP to ~20 MFLOP; runtime is bound by the
// 6.46 MB output write (+8.2 MB input read) -> ~0.64 us at 23.3 TB/s.

typedef __attribute__((ext_vector_type(2))) float v2f;
typedef __attribute__((ext_vector_type(8))) float v8f;

#define Hdim 512
#define Vdim 5
#define Bdim 8
#define Tdim 400
#define U1dim 101

#define TROWS (Bdim * Tdim)    // 3200
#define PROWS (Bdim * U1dim)   // 808
#define TTILES ((TROWS + 15) / 16)            // 200
#define PTILES ((PROWS + 15) / 16)            // 51
#define NTILES (TTILES + PTILES)              // 251

// ---------------------------------------------------------------------------
// Phase 1: tiny row-block GEMMs via V_WMMA_F32_16X16X4_F32.
// One wave32 per 16-row tile; N padded 5 -> 16; K loop = 512/4 = 128 WMMAs.
// Bias folded into the trans-GEMM accumulator init.
//
// No predicated loads: OOB A-rows / B-columns are CLAMPED to a valid address;
// the garbage they produce lands only in D rows/columns that the store guards
// already skip.  Inner loop = 2x global_load_b64 + v_wmma, fully pipelineable.
// ---------------------------------------------------------------------------
__global__ __launch_bounds__(256) void rowgemm_wmma(
    const float* __restrict__ trans,   // [3200, 512]
    const float* __restrict__ preds,   // [ 808, 512]
    const float* __restrict__ W,       // [   5, 512]
    const float* __restrict__ bias,    // [5]
    float* __restrict__ TW,            // [3200, 5]
    float* __restrict__ PW)            // [ 808, 5]
{
    const int wave = threadIdx.x >> 5;             // 0..7
    const int lane = threadIdx.x & 31;
    const int tile = blockIdx.x * 8 + wave;        // 0..255
    if (tile >= NTILES) return;                    // wave-uniform exit

    const bool  isTrans = (tile < TTILES);
    const float* A   = isTrans ? trans : preds;
    float*       Dst = isTrans ? TW    : PW;
    const int    rows = isTrans ? TROWS : PROWS;
    const int    row0 = (isTrans ? tile : tile - TTILES) * 16;

    const int half = lane >> 4;      // 0 or 1 (selects K pair per ISA layout)
    const int l15  = lane & 15;      // A: row within tile; B/C: column N

    // A-matrix 16x4 f32 layout: lanes 0-15 hold K={k,k+1}, lanes 16-31 K={k+2,k+3}
    const int arow = min(row0 + l15, rows - 1);          // clamp, no predication
    // B-matrix 4x16 f32 (mirrored layout): B[k][n] = W[n][k], n = l15
    const int brow = min(l15, Vdim - 1);                 // clamp, no predication

    const float* __restrict__ ap = A + arow * Hdim + 2 * half;
    const float* __restrict__ bp = W + brow * Hdim + 2 * half;

    // C/D 16x16 f32: VGPR r -> row (row0 + r + 8*half), column l15.
    v8f c;
    const float cinit = isTrans ? bias[brow] : 0.0f;
#pragma unroll
    for (int r = 0; r < 8; ++r) c[r] = cinit;

#pragma unroll 8
    for (int k = 0; k < Hdim; k += 4) {
        const v2f a = *(const v2f*)(ap + k);   // global_load_b64
        const v2f b = *(const v2f*)(bp + k);   // global_load_b64
        // D = A(16x4) x B(4x16) + C ; emits v_wmma_f32_16x16x4_f32
        c = __builtin_amdgcn_wmma_f32_16x16x4_f32(
                /*neg_a=*/false, a, /*neg_b=*/false, b,
                /*c_mod=*/(short)0, c, /*reuse_a=*/false, /*reuse_b=*/false);
    }

    if (l15 < Vdim) {
        const int n = l15;
#pragma unroll
        for (int r = 0; r < 8; ++r) {
            const int row = row0 + r + 8 * half;
            if (row < rows) Dst[row * Vdim + n] = c[r];
        }
    }
}

// ---------------------------------------------------------------------------
// Phase 2: bandwidth-bound fan-out.  One block per (b,t); writes the
// contiguous 505-float [u,v] slab.  TW/PW reads are cache-resident
// (64 KB + 16 KB << 192 MB L2); output stores are contiguous, coalesced,
// and NON-TEMPORAL (written once, never re-read on device).
// LeakyReLU(0.1): x>=0 ? x : 0.1x  ==  max(x, 0.1x).
// ---------------------------------------------------------------------------
__global__ __launch_bounds__(256) void joint_write(
    const float* __restrict__ TW,      // [3200, 5]  (bias included)
    const float* __restrict__ PW,      // [ 808, 5]
    float* __restrict__ out)           // [3200, 505]
{
    const int bt = blockIdx.x;                   // 0..3199  (= b*T + t)
    const int bb = bt / Tdim;                    // batch index
    const float* __restrict__ pw = PW + bb * (U1dim * Vdim);  // 505 floats
    const float* __restrict__ tw = TW + bt * Vdim;            // 5 floats
    float* __restrict__ o = out + (size_t)bt * (U1dim * Vdim);

    for (int j = threadIdx.x; j < U1dim * Vdim; j += 256) {
        const int v = j % Vdim;
        const float x = tw[v] + pw[j];
        __builtin_nontemporal_store(fmaxf(x, 0.1f * x), o + j);
    }
}

extern "C" void kernel_launch(void* const* d_in, const int* in_sizes, int n_in,
                              void* d_out, int out_size, void* d_ws, size_t ws_size,
                              hipStream_t stream) {
    const float* trans = (const float*)d_in[0];   // [8,400,512]
    const float* preds = (const float*)d_in[1];   // [8,101,512]
    const float* W     = (const float*)d_in[2];   // [5,512]
    const float* bias  = (const float*)d_in[3];   // [5]
    float* out = (float*)d_out;                   // [8,400,101,5]

    float* TW = (float*)d_ws;                     // 3200*5 floats = 64 KB
    float* PW = TW + TROWS * Vdim;                //  808*5 floats = 16 KB

    // 251 tiles, 8 waves/block -> 32 blocks
    rowgemm_wmma<<<(NTILES + 7) / 8, 256, 0, stream>>>(trans, preds, W, bias, TW, PW);
    // one block per (b,t) slab
    joint_write<<<TROWS, 256, 0, stream>>>(TW, PW, out);
}